// FSNeuron_80058190397910
// MI455X (gfx1250) — compile-verified
//
#include <hip/hip_runtime.h>

// FS-neuron K=16 scan, elementwise. Roofline: 134 MB traffic ~ 5.8us at
// 23.3 TB/s; VALU is the co-limiter, so the step is compressed to
// 3 VALU ops/element using gfx1250 V_PK_FMA_F32 (VOP3P packed f32) with
// op_sel broadcast of the (d_t, -h_t) constant pair.

typedef __attribute__((ext_vector_type(2))) float v2f;

constexpr int K_STEPS = 16;
constexpr int TPB     = 256;            // 8 wave32 per block
constexpr int UNROLL  = 4;              // float4 per thread
constexpr int PAIRS   = UNROLL * 2;     // 8 independent v2f element pairs

// acc.{lo,hi} += dh.lo * zz.{lo,hi}   (broadcast low half of src0)
__device__ __forceinline__ void pk_fma_bcast_lo(v2f& acc, v2f dh, v2f zz) {
    asm("v_pk_fma_f32 %0, %1, %2, %0 op_sel_hi:[0,1,1]"
        : "+v"(acc) : "v"(dh), "v"(zz));
}
// acc.{lo,hi} += dh.hi * zz.{lo,hi}   (broadcast high half of src0)
__device__ __forceinline__ void pk_fma_bcast_hi(v2f& acc, v2f dh, v2f zz) {
    asm("v_pk_fma_f32 %0, %1, %2, %0 op_sel:[1,0,0] op_sel_hi:[1,1,1]"
        : "+v"(acc) : "v"(dh), "v"(zz));
}

__global__ __launch_bounds__(TPB) void fs_neuron_kernel(
    const float* __restrict__ x,
    const float* __restrict__ h,
    const float* __restrict__ d,
    const float* __restrict__ T,
    float* __restrict__ out,
    int n)
{
    const int n4 = n >> 2;
    const float4* __restrict__ x4 = reinterpret_cast<const float4*>(x);
    float4*       __restrict__ o4 = reinterpret_cast<float4*>(out);

    const size_t tile = (size_t)blockIdx.x * (TPB * UNROLL);
    const size_t base = tile + threadIdx.x;

    if (tile + (size_t)(TPB * UNROLL) <= (size_t)n4) {
        // ---------------- fast path: full tile, zero predication ----------------
        // Step constants: T in SGPRs (v_cmp scalar operand), (d, -h) as one
        // VGPR pair per step, broadcast via op_sel in the pk_fma.
        float Tq[K_STEPS];
        v2f   dh[K_STEPS];
#pragma unroll
        for (int t = 0; t < K_STEPS; ++t) {
            Tq[t] = T[t];
            v2f p; p.x = d[t]; p.y = -h[t];
            dh[t] = p;
        }

        v2f vv[PAIRS], oo[PAIRS];
#pragma unroll
        for (int u = 0; u < UNROLL; ++u) {
            const float4 xv = x4[base + (size_t)u * TPB];     // global_load_b128
            v2f a; a.x = xv.x; a.y = xv.y;
            v2f b; b.x = xv.z; b.y = xv.w;
            vv[2 * u]     = a;
            vv[2 * u + 1] = b;
            v2f z0; z0.x = 0.f; z0.y = 0.f;
            oo[2 * u]     = z0;
            oo[2 * u + 1] = z0;
        }

#pragma unroll
        for (int t = 0; t < K_STEPS; ++t) {
            const float Tt = Tq[t];
            const v2f  dht = dh[t];
#pragma unroll
            for (int j = 0; j < PAIRS; ++j) {
                v2f zz;                                        // (z0, z1) built by 2 cndmasks
                zz.x = (vv[j].x > Tt) ? 1.0f : 0.0f;           // v_cmp_gt_f32 + v_cndmask
                zz.y = (vv[j].y > Tt) ? 1.0f : 0.0f;
                pk_fma_bcast_lo(oo[j], dht, zz);               // out += d_t  * z  (2 lanes)
                pk_fma_bcast_hi(vv[j], dht, zz);               // v   += -h_t * z  (2 lanes)
            }
        }

#pragma unroll
        for (int u = 0; u < UNROLL; ++u) {
            float4 r;
            r.x = oo[2 * u].x;     r.y = oo[2 * u].y;
            r.z = oo[2 * u + 1].x; r.w = oo[2 * u + 1].y;
            o4[base + (size_t)u * TPB] = r;                    // global_store_b128
        }
    } else {
        // ---------------- edge path: last partial tile, compact & rolled ----------------
        for (int u = 0; u < UNROLL; ++u) {
            const size_t i = base + (size_t)u * TPB;
            if (i < (size_t)n4) {
                const float4 xv = x4[i];
                float vv[4] = {xv.x, xv.y, xv.z, xv.w};
                float oo[4] = {0.f, 0.f, 0.f, 0.f};
#pragma clang loop unroll(disable)
                for (int t = 0; t < K_STEPS; ++t) {
                    const float Tt = T[t], dt = d[t], ht = h[t];
#pragma unroll
                    for (int c = 0; c < 4; ++c) {
                        const float z = (vv[c] > Tt) ? 1.0f : 0.0f;
                        oo[c] = __builtin_fmaf(z,  dt, oo[c]);
                        vv[c] = __builtin_fmaf(z, -ht, vv[c]);
                    }
                }
                float4 r; r.x = oo[0]; r.y = oo[1]; r.z = oo[2]; r.w = oo[3];
                o4[i] = r;
            }
        }
    }

    // ---------------- tail: n % 4 scalars (absent for the reference shape) ----------------
    const int rem = n & 3;
    if (rem) {
        const int gid = (int)(blockIdx.x * TPB + threadIdx.x);
        if (gid < rem) {
            const int i = (n4 << 2) + gid;
            float vv = x[i], oo = 0.f;
#pragma clang loop unroll(disable)
            for (int t = 0; t < K_STEPS; ++t) {
                const float z = (vv > T[t]) ? 1.0f : 0.0f;
                oo = __builtin_fmaf(z,  d[t], oo);
                vv = __builtin_fmaf(z, -h[t], vv);
            }
            out[i] = oo;
        }
    }
}

extern "C" void kernel_launch(void* const* d_in, const int* in_sizes, int n_in,
                              void* d_out, int out_size, void* d_ws, size_t ws_size,
                              hipStream_t stream) {
    const float* x = (const float*)d_in[0];
    const float* h = (const float*)d_in[1];
    const float* d = (const float*)d_in[2];
    const float* T = (const float*)d_in[3];
    float* out = (float*)d_out;

    const int n  = in_sizes[0];          // B*L*D = 16,777,216
    const int n4 = n >> 2;
    const int per_block = TPB * UNROLL;  // 1024 float4 per block
    int blocks = (n4 + per_block - 1) / per_block;
    if (blocks < 1) blocks = 1;

    fs_neuron_kernel<<<blocks, TPB, 0, stream>>>(x, h, d, T, out, n);
}